// EdgeEncoder_14190571946553
// MI455X (gfx1250) — compile-verified
//
#include <hip/hip_runtime.h>
#include <hip/hip_bf16.h>

// ---------------- static config (mirrors reference) ----------------
#define NE        131072
#define DDIM      256
#define NT_BASE   9
#define NTYPES    14
#define NSRC      5
#define PMAX      4
#define MCHUNK    128          // edges per block in main kernel
#define HSTRIDE   260          // LDS row stride (floats) -> conflict-free A reads

typedef float v2f __attribute__((ext_vector_type(2)));
typedef float v8f __attribute__((ext_vector_type(8)));

__constant__ int   c_BASEMAP[NTYPES] = {0,0,0,1,1,1,2,2,3,4,5,6,7,8};
__constant__ int   c_PCOUNT[NT_BASE] = {2,2,1,1,1,1,3,2,4};
__constant__ float c_SCALES[NT_BASE][PMAX] = {
    {1.0f, 1e-6f, 1.0f, 1.0f},   // nmos  (m,w)
    {1.0f, 1e-6f, 1.0f, 1.0f},   // pmos
    {1.0f, 1.0f,  1.0f, 1.0f},   // balun (rout)
    {1000.0f, 1.0f, 1.0f, 1.0f}, // resistor (r)
    {1e-12f, 1.0f, 1.0f, 1.0f},  // capacitor (c)
    {1e-9f, 1.0f, 1.0f, 1.0f},   // inductor (l)
    {1.0f, 1.0f, 1.0f, 1.0f},    // vsource (dc,mag,phase)
    {1e-3f, 1e-3f, 1.0f, 1.0f},  // isource (dc,mag)
    {1.0f, 1.0f, 1e9f, 1.0f}     // port (dbm,dc,freq,num)
};

__device__ __forceinline__ v8f wmma4(v2f a, v2f b, v8f c) {
    // D = A(16x4 f32) * B(4x16 f32) + C ; fp32 accumulate, matches reference precision
    return __builtin_amdgcn_wmma_f32_16x16x4_f32(false, a, false, b, (short)0, c,
                                                 false, false);
}

// Packed B layout so the main GEMM does perfectly coalesced float4 loads:
// for k-pair-group kkp (8 k's) / n-tile nt, lane L holds
//   {B[8kkp+2h][n], B[8kkp+2h+1][n], B[8kkp+4+2h][n], B[8kkp+5+2h][n]}
// with h = L>>4, n = nt*16 + (L&15).
__device__ __forceinline__ int packed_b_index(int k, int n) {
    int kkp  = k >> 3;
    int rem  = k & 7;
    int h2   = (rem >> 1) & 1;
    int q    = ((rem & 4) >> 1) | (rem & 1);
    int lane = (h2 << 4) | (n & 15);
    return ((((n >> 4) * 32 + kkp) * 32 + lane) << 2) + q;
}

// ---------------- pipeline kernels ----------------

__global__ void ee_zero(int* counts, int* cursor) {
    int i = threadIdx.x;
    if (i < 16) { counts[i] = 0; cursor[i] = 0; }
}

__global__ void ee_hist(const int* __restrict__ type_ids, int* __restrict__ counts, int n) {
    int i = blockIdx.x * blockDim.x + threadIdx.x;
    if (i < n) atomicAdd(&counts[c_BASEMAP[type_ids[i]]], 1);
}

__global__ void ee_scan(const int* __restrict__ counts, int* __restrict__ offsets) {
    if (threadIdx.x == 0 && blockIdx.x == 0) {
        int acc = 0;
        for (int t = 0; t < NT_BASE; ++t) { offsets[t] = acc; acc += counts[t]; }
    }
}

__global__ void ee_scatter(const int* __restrict__ type_ids,
                           const int* __restrict__ offsets, int* __restrict__ cursor,
                           int* __restrict__ sorted, int n) {
    int i = blockIdx.x * blockDim.x + threadIdx.x;
    if (i < n) {
        int t = c_BASEMAP[type_ids[i]];
        int pos = offsets[t] + atomicAdd(&cursor[t], 1);
        sorted[pos] = i;
    }
}

// CTS[ti][si][j] = bf[j] + type_embed[ti]@Wf_top + source_embed[si]@Wf_mid
//                + b2[base(ti)]@Wf_bot
__global__ void ee_cts(const float* __restrict__ te, const float* __restrict__ se,
                       const float* __restrict__ b2, const float* __restrict__ Wf,
                       const float* __restrict__ bf, float* __restrict__ CTS) {
    int blk = blockIdx.x;            // 0..69
    int ti = blk / NSRC, si = blk % NSRC;
    int j = threadIdx.x;
    int t = c_BASEMAP[ti];
    float acc = bf[j];
    for (int k = 0; k < DDIM; ++k) acc += te[ti * DDIM + k] * Wf[k * DDIM + j];
    for (int k = 0; k < DDIM; ++k) acc += se[si * DDIM + k] * Wf[(DDIM + k) * DDIM + j];
    for (int k = 0; k < DDIM; ++k) acc += b2[t * DDIM + k] * Wf[(2 * DDIM + k) * DDIM + j];
    CTS[blk * DDIM + j] = acc;
}

// W2f[t] = W2[t] (256x256) @ Wf_bot (256x256), written in packed-WMMA-B layout.
// One wave (32 threads) per 16x16 output tile; 9*256 blocks.
__global__ void ee_pack_w2f(const float* __restrict__ W2, const float* __restrict__ Wfb,
                            float* __restrict__ PB) {
    const int b = blockIdx.x;
    const int t = b >> 8;
    const int tile = b & 255;
    const int mt = tile >> 4, ntile = tile & 15;
    const int lane = threadIdx.x & 31;
    const int ln = lane & 15, hh = lane >> 4;
    const int m = mt * 16 + ln;
    const int n = ntile * 16 + ln;
    const float* A = W2 + ((size_t)t * DDIM + m) * DDIM;
    v8f acc = {};
    for (int c = 0; c < 64; ++c) {
        int k0 = 4 * c + 2 * hh;
        v2f a = *(const v2f*)(A + k0);
        v2f bb;
        bb.x = Wfb[k0 * DDIM + n];
        bb.y = Wfb[(k0 + 1) * DDIM + n];
        acc = wmma4(a, bb, acc);
    }
    float* PBt = PB + (size_t)t * (DDIM * DDIM);
#pragma unroll
    for (int r = 0; r < 8; ++r) {
        int kg = mt * 16 + r + 8 * hh;   // GEMM M index == main-GEMM K index
        PBt[packed_b_index(kg, n)] = acc[r];
    }
}

// Main grouped GEMM: out[e] = relu( h[e] @ W2f[t] + CTS[tid*5+sid] )
// h[e] = relu( x[e] @ W1[t] + b1[t] ), x = masked/scaled params.
// Block: 256 threads (8 waves), 128 edges of one base type, N=256 full.
__global__ void ee_main(const int* __restrict__ sorted, const int* __restrict__ counts,
                        const int* __restrict__ offsets, const int* __restrict__ type_ids,
                        const int* __restrict__ source_ids, const float* __restrict__ params,
                        const float* __restrict__ W1, const float* __restrict__ b1,
                        const float* __restrict__ PB, const float* __restrict__ CTS,
                        float* __restrict__ out, int bpt) {
    extern __shared__ char smem[];
    float* Hs   = (float*)smem;                 // [128][HSTRIDE]
    float* Xs   = Hs + MCHUNK * HSTRIDE;        // [128][4]
    int*   rowE = (int*)(Xs + MCHUNK * PMAX);   // [128]
    int*   rowC = rowE + MCHUNK;                // [128]

    const int t     = blockIdx.x / bpt;
    const int chunk = blockIdx.x % bpt;
    const int cnt   = counts[t];
    const int rowStart = chunk * MCHUNK;
    if (rowStart >= cnt) return;
    const int off  = offsets[t];
    const int tidx = threadIdx.x;

    // phase 0: stage edge indices, combined bias index, scaled params
    if (tidx < MCHUNK) {
        int gr = rowStart + tidx;
        bool valid = gr < cnt;
        int e = valid ? sorted[off + gr] : 0;
        rowE[tidx] = e;
        rowC[tidx] = type_ids[e] * NSRC + source_ids[e];
        int pc = c_PCOUNT[t];
#pragma unroll
        for (int p = 0; p < PMAX; ++p) {
            float v = 0.0f;
            if (valid && p < pc) v = params[(size_t)e * PMAX + p] / c_SCALES[t][p];
            Xs[tidx * PMAX + p] = v;
        }
    }
    __syncthreads();

    // phase 1: h = relu(x @ W1[t] + b1[t]) -> LDS (thread j owns column j)
    {
        const int j = tidx;
        float w0 = W1[(t * PMAX + 0) * DDIM + j];
        float w1 = W1[(t * PMAX + 1) * DDIM + j];
        float w2 = W1[(t * PMAX + 2) * DDIM + j];
        float w3 = W1[(t * PMAX + 3) * DDIM + j];
        float bb = b1[t * DDIM + j];
        for (int r = 0; r < MCHUNK; ++r) {
            float h = bb + Xs[r * PMAX + 0] * w0 + Xs[r * PMAX + 1] * w1
                         + Xs[r * PMAX + 2] * w2 + Xs[r * PMAX + 3] * w3;
            Hs[r * HSTRIDE + j] = fmaxf(h, 0.0f);
        }
    }
    __syncthreads();

    // phase 2: WMMA GEMM, wave w owns M-tile w (16 rows) x all 16 N-tiles
    const int w = tidx >> 5, lane = tidx & 31;
    const int ln = lane & 15, hh = lane >> 4;
    const int mrow = w * 16 + ln;
    const float* PBt  = PB + (size_t)t * (DDIM * DDIM);
    const float* Arow = Hs + mrow * HSTRIDE;

    v8f acc[16];
    v8f z = {};
#pragma unroll
    for (int nt = 0; nt < 16; ++nt) acc[nt] = z;

    for (int kkp = 0; kkp < 32; ++kkp) {           // 8 K values per iter
        v2f a0 = *(const v2f*)(Arow + 8 * kkp + 2 * hh);
        v2f a1 = *(const v2f*)(Arow + 8 * kkp + 4 + 2 * hh);
#pragma unroll
        for (int nt = 0; nt < 16; ++nt) {
            const float4 f = *(const float4*)(PBt + (((nt * 32 + kkp) * 32 + lane) << 2));
            v2f b0; b0.x = f.x; b0.y = f.y;
            v2f b1v; b1v.x = f.z; b1v.y = f.w;
            acc[nt] = wmma4(a0, b0, acc[nt]);
            acc[nt] = wmma4(a1, b1v, acc[nt]);
        }
    }

    // epilogue: add fused bias table, relu, store
#pragma unroll
    for (int r = 0; r < 8; ++r) {
        int mloc = w * 16 + r + 8 * hh;
        int gr = rowStart + mloc;
        if (gr >= cnt) continue;
        int e = rowE[mloc];
        const float* cts = CTS + (size_t)rowC[mloc] * DDIM;
        float* orow = out + (size_t)e * DDIM;
#pragma unroll
        for (int nt = 0; nt < 16; ++nt) {
            int n = nt * 16 + ln;
            orow[n] = fmaxf(acc[nt][r] + cts[n], 0.0f);
        }
    }
}

// ---------------- launcher ----------------
extern "C" void kernel_launch(void* const* d_in, const int* in_sizes, int n_in,
                              void* d_out, int out_size, void* d_ws, size_t ws_size,
                              hipStream_t stream) {
    (void)n_in; (void)out_size; (void)ws_size;
    const int*   type_ids   = (const int*)d_in[0];
    const int*   source_ids = (const int*)d_in[1];
    const float* params     = (const float*)d_in[2];
    const float* type_embed = (const float*)d_in[3];
    const float* src_embed  = (const float*)d_in[4];
    const float* W1         = (const float*)d_in[5];
    const float* b1         = (const float*)d_in[6];
    const float* W2         = (const float*)d_in[7];
    const float* b2         = (const float*)d_in[8];
    const float* Wf         = (const float*)d_in[9];
    const float* bf         = (const float*)d_in[10];
    float*       out        = (float*)d_out;

    const int E = in_sizes[0];

    char* ws = (char*)d_ws;
    int*   counts  = (int*)(ws + 0);
    int*   offsets = (int*)(ws + 64);
    int*   cursor  = (int*)(ws + 128);
    int*   sorted  = (int*)(ws + 1024);
    float* CTS     = (float*)(ws + 1024 + (size_t)E * sizeof(int));               // 70*256 f32
    float* PB      = (float*)(ws + 1024 + (size_t)E * sizeof(int)
                                   + (size_t)NTYPES * NSRC * DDIM * sizeof(float));// 9*256*256 f32

    ee_zero<<<1, 64, 0, stream>>>(counts, cursor);
    ee_hist<<<(E + 255) / 256, 256, 0, stream>>>(type_ids, counts, E);
    ee_scan<<<1, 32, 0, stream>>>(counts, offsets);
    ee_scatter<<<(E + 255) / 256, 256, 0, stream>>>(type_ids, offsets, cursor, sorted, E);
    ee_cts<<<NTYPES * NSRC, DDIM, 0, stream>>>(type_embed, src_embed, b2, Wf, bf, CTS);
    ee_pack_w2f<<<NT_BASE * 256, 32, 0, stream>>>(W2, Wf + 2 * DDIM * DDIM, PB);

    const int bpt = (E + MCHUNK - 1) / MCHUNK;     // worst-case tiles per type
    const size_t smem = (MCHUNK * HSTRIDE + MCHUNK * PMAX) * sizeof(float)
                      + 2 * MCHUNK * sizeof(int);
    ee_main<<<NT_BASE * bpt, 256, smem, stream>>>(sorted, counts, offsets, type_ids,
                                                  source_ids, params, W1, b1, PB, CTS,
                                                  out, bpt);
}